// PoseGaussianLoss_45062796869822
// MI455X (gfx1250) — compile-verified
//
#include <hip/hip_runtime.h>
#include <math.h>

typedef __attribute__((ext_vector_type(2))) float v2f;
typedef __attribute__((ext_vector_type(8))) float v8f;

#define NBLOCKS   512
#define NTHREADS  256
#define B_TOTAL   2000000
#define NGROUPS   (B_TOTAL / 4)   // 4 items per thread-iteration

// Full 32-lane wave sum via V_WMMA_F32_16X16X4_F32.
// A layout (16x4 f32): VGPR0 lanes0-15 -> K=0, lanes16-31 -> K=2;
//                      VGPR1 lanes0-15 -> K=1, lanes16-31 -> K=3.
// With B == all-ones, D[M][N] = sum_K A[M][K], independent of B's layout.
// Lane carries (partial, 0) -> row M of D == p_M + p_{M+16}. Summing the 8
// C/D VGPRs per lane covers M=0..7 (lanes<16) / M=8..15 (lanes>=16); one
// shfl_xor(16) merges the halves -> total of all 32 partials, exact f32.
__device__ __forceinline__ float wave_sum_wmma(float x) {
    v2f a; a[0] = x;    a[1] = 0.0f;
    v2f b; b[0] = 1.0f; b[1] = 1.0f;
    v8f c = {};
    c = __builtin_amdgcn_wmma_f32_16x16x4_f32(
            /*neg_a=*/false, a, /*neg_b=*/false, b,
            /*c_mod=*/(short)0, c, /*reuse_a=*/false, /*reuse_b=*/false);
    float s = ((c[0] + c[1]) + (c[2] + c[3])) + ((c[4] + c[5]) + (c[6] + c[7]));
    s += __shfl_xor(s, 16, 32);
    return s;
}

// loss_i = e^T adj(S) e / det(S) + log(det(S) + 1), S = 3x3 row-major
__device__ __forceinline__ float item_loss(float e0, float e1, float e2,
                                           const float* s) {
    float a = s[0], b = s[1], c = s[2];
    float d = s[3], e = s[4], f = s[5];
    float g = s[6], h = s[7], i = s[8];
    float A00 = e*i - f*h;
    float A01 = c*h - b*i;
    float A02 = b*f - c*e;
    float A10 = f*g - d*i;
    float A11 = a*i - c*g;
    float A12 = c*d - a*f;
    float A20 = d*h - e*g;
    float A21 = b*g - a*h;
    float A22 = a*e - b*d;
    float det = a*A00 + b*A10 + c*A20;
    float q0 = A00*e0 + A01*e1 + A02*e2;
    float q1 = A10*e0 + A11*e1 + A12*e2;
    float q2 = A20*e0 + A21*e1 + A22*e2;
    float quad = e0*q0 + e1*q1 + e2*q2;
    return quad / det + logf(det + 1.0f);
}

__global__ void __launch_bounds__(NTHREADS)
pose_loss_partial(const float* __restrict__ p_delta,
                  const float* __restrict__ p_sigma,
                  const float* __restrict__ pose_gt,
                  float* __restrict__ partials) {
    const float4* pd4 = (const float4*)p_delta;
    const float4* pg4 = (const float4*)pose_gt;
    const float4* ps4 = (const float4*)p_sigma;

    float acc = 0.0f;
    int tid    = blockIdx.x * blockDim.x + threadIdx.x;
    int stride = gridDim.x * blockDim.x;

    for (int grp = tid; grp < NGROUPS; grp += stride) {
        float dd[12], gg[12], ss[36];
        #pragma unroll
        for (int k = 0; k < 3; ++k) {           // 12 deltas + 12 gts, b128 loads
            float4 t = pd4[3 * grp + k];
            dd[4*k+0] = t.x; dd[4*k+1] = t.y; dd[4*k+2] = t.z; dd[4*k+3] = t.w;
            float4 u = pg4[3 * grp + k];
            gg[4*k+0] = u.x; gg[4*k+1] = u.y; gg[4*k+2] = u.z; gg[4*k+3] = u.w;
        }
        #pragma unroll
        for (int k = 0; k < 9; ++k) {           // 36 sigma floats, b128 loads
            float4 t = ps4[9 * grp + k];
            ss[4*k+0] = t.x; ss[4*k+1] = t.y; ss[4*k+2] = t.z; ss[4*k+3] = t.w;
        }
        #pragma unroll
        for (int it = 0; it < 4; ++it) {
            acc += item_loss(dd[3*it+0] - gg[3*it+0],
                             dd[3*it+1] - gg[3*it+1],
                             dd[3*it+2] - gg[3*it+2],
                             ss + 9*it);
        }
    }

    // Deterministic block reduction: WMMA wave sums -> LDS -> thread 0.
    __shared__ float lsum[NTHREADS / 32];
    float w = wave_sum_wmma(acc);
    int lane = threadIdx.x & 31;
    int wid  = threadIdx.x >> 5;
    if (lane == 0) lsum[wid] = w;
    __syncthreads();
    if (threadIdx.x == 0) {
        float t = 0.0f;
        #pragma unroll
        for (int k = 0; k < NTHREADS / 32; ++k) t += lsum[k];
        partials[blockIdx.x] = t;
    }
}

__global__ void __launch_bounds__(256)
pose_loss_final(const float* __restrict__ partials, float* __restrict__ out) {
    // 512 partials, 256 threads: two loads each, then WMMA wave sums.
    float x = partials[threadIdx.x] + partials[threadIdx.x + 256];
    __shared__ float lsum[8];
    float w = wave_sum_wmma(x);
    int lane = threadIdx.x & 31;
    int wid  = threadIdx.x >> 5;
    if (lane == 0) lsum[wid] = w;
    __syncthreads();
    if (threadIdx.x == 0) {
        float t = 0.0f;
        #pragma unroll
        for (int k = 0; k < 8; ++k) t += lsum[k];
        out[0] = t / (float)B_TOTAL;
    }
}

extern "C" void kernel_launch(void* const* d_in, const int* in_sizes, int n_in,
                              void* d_out, int out_size, void* d_ws, size_t ws_size,
                              hipStream_t stream) {
    const float* p_delta = (const float*)d_in[0];
    const float* p_sigma = (const float*)d_in[1];
    const float* pose_gt = (const float*)d_in[2];
    float* out      = (float*)d_out;
    float* partials = (float*)d_ws;   // 512 floats = 2 KB scratch

    pose_loss_partial<<<NBLOCKS, NTHREADS, 0, stream>>>(p_delta, p_sigma,
                                                        pose_gt, partials);
    pose_loss_final<<<1, 256, 0, stream>>>(partials, out);
}